// YOLOV3PostProcess_15719580304017
// MI455X (gfx1250) — compile-verified
//
#include <hip/hip_runtime.h>
#include <hip/hip_bf16.h>

// ---------------------------------------------------------------------------
// YOLOv3 post-process for MI455X (gfx1250, wave32).
// decode -> histogram top-k (exact, bisection-refined) -> greedy NMS.
// CDNA5 paths: TDM (tensor_store_from_lds / tensor_load_to_lds) moves the
// 80KB per-image candidate tile (4 box planes + score plane) between LDS and
// global; completion fenced with s_wait_tensorcnt.
// ---------------------------------------------------------------------------

#define NUM_CLASSES 80
#define NANCH       22743            // 3*(19^2 + 38^2 + 76^2)
#define NA0         1083             // level-0 anchors
#define NB01        5415             // level-0+1 anchors
#define HW0         361
#define HW1         1444
#define HW2         5776
#define PRE_K       4096
#define TOP_K       200
#define BOUND_CAP   8192
#define NEGV        (-1.0e9f)
#define SCORE_TH    0.01f
#define NMS_TH      0.45f
#define NBIN        4096
#define BS          8
#define TILE_NE     (5 * PRE_K)      // 4 box planes + 1 score plane (dwords)

struct Ins {
    const float* in[3];
    const float* aw[3];
    const float* ah[3];
    const float* gx[3];
    const float* gy[3];
};

__device__ __forceinline__ float sigm(float x) { return 1.0f / (1.0f + expf(-x)); }

__device__ __forceinline__ void resolveAnchor(int g, int& lvl, int& local, int& hw, float& stride) {
    if (g < NA0)       { lvl = 0; local = g;        hw = HW0; stride = 32.0f; }
    else if (g < NB01) { lvl = 1; local = g - NA0;  hw = HW1; stride = 16.0f; }
    else               { lvl = 2; local = g - NB01; hw = HW2; stride = 8.0f;  }
}

__device__ __forceinline__ int scoreBin(float s) {
    int b = (int)(s * (float)NBIN);
    return b > (NBIN - 1) ? (NBIN - 1) : b;
}

// ---- Tensor DMA descriptor (D#) for a flat 1-D dword tile --------------------
typedef unsigned int u32x4v __attribute__((ext_vector_type(4)));
typedef int          i32x8v __attribute__((ext_vector_type(8)));
typedef int          i32x4v __attribute__((ext_vector_type(4)));

__device__ __forceinline__ void make_tdm_desc(unsigned int ldsAddr, unsigned long long ga,
                                              int ne, u32x4v& g0, i32x8v& g1) {
    g0[0] = 1u;                                               // count=1 (valid D#)
    g0[1] = ldsAddr;                                          // lds_addr @ [63:32]
    g0[2] = (unsigned int)(ga & 0xFFFFFFFFull);               // global_addr lo
    g0[3] = (unsigned int)((ga >> 32) & 0x1FFFFFFu) | 0x80000000u; // hi | type=2
    g1[0] = (2 << 16);                                        // data_size=4B, wg_mask=0
    g1[1] = (int)((ne & 0xFFFF) << 16);                       // tensor_dim0[15:0] @ [63:48]
    g1[2] = (int)(((ne >> 16) & 0xFFFF) | (1 << 16));         // tensor_dim0 hi | tensor_dim1=1
    g1[3] = (int)((ne & 0xFFFF) << 16);                       // tile_dim0 @ [127:112]
    g1[4] = 1;                                                // tile_dim1=1, tile_dim2=0
    g1[5] = ne;                                               // tensor_dim0_stride lo
    g1[6] = (int)((ne & 0xFFFF) << 16);                       // tensor_dim1_stride lo @ [223:208]
    g1[7] = 0;
}

// ---------------------------------------------------------------------------
__global__ __launch_bounds__(256) void k_init(unsigned int* hist, unsigned int* ctrA,
                                              unsigned int* ctrB) {
    int i = blockIdx.x * blockDim.x + threadIdx.x;
    if (i < BS * NBIN) hist[i] = 0u;
    if (i < BS) { ctrA[i] = 0u; ctrB[i] = 0u; }
}

// ---------------------------------------------------------------------------
// Pass 1: decode boxes (corner form, stride-scaled) + per-image score histogram.
__global__ __launch_bounds__(256) void k_decode_hist(Ins ins, float4* boxesWS,
                                                     unsigned int* hist) {
    __shared__ unsigned int lh[NBIN];
    const int b = blockIdx.y;
    const int g = blockIdx.x * blockDim.x + threadIdx.x;
    for (int i = threadIdx.x; i < NBIN; i += blockDim.x) lh[i] = 0u;
    __syncthreads();

    if (g < NANCH) {
        int lvl, local, hw; float stride;
        resolveAnchor(g, lvl, local, hw, stride);
        const int a   = local / hw;
        const int rem = local - a * hw;
        const float* inp = ins.in[lvl];
        const int base = ((b * 3 + a) * (NUM_CLASSES + 5)) * hw + rem;
        const int gi   = (b * 3 + a) * hw + rem;

        const float tx = inp[base];
        const float ty = inp[base + hw];
        const float tw = inp[base + 2 * hw];
        const float th = inp[base + 3 * hw];
        const float tc = inp[base + 4 * hw];

        const float cx = (sigm(tx) + ins.gx[lvl][gi]) * stride;
        const float cy = (sigm(ty) + ins.gy[lvl][gi]) * stride;
        const float bw = expf(tw) * ins.aw[lvl][gi] * stride;
        const float bh = expf(th) * ins.ah[lvl][gi] * stride;
        boxesWS[b * NANCH + g] =
            make_float4(cx - bw * 0.5f, cy - bh * 0.5f, cx + bw * 0.5f, cy + bh * 0.5f);

        const float conf = sigm(tc);
        for (int c = 0; c < NUM_CLASSES; ++c) {
            const float s = conf * sigm(inp[base + (5 + c) * hw]);
            if (s > SCORE_TH) atomicAdd(&lh[scoreBin(s)], 1u);
        }
    }
    __syncthreads();
    for (int i = threadIdx.x; i < NBIN; i += blockDim.x)
        if (lh[i]) atomicAdd(&hist[b * NBIN + i], lh[i]);
}

// ---------------------------------------------------------------------------
// Find cutoff bin per image: largest bin B* such that count(bins >= B*) >= PRE_K.
__global__ __launch_bounds__(512) void k_scan(const unsigned int* hist, int* meta) {
    __shared__ unsigned int part[512];
    __shared__ int cutBin;
    const int b = blockIdx.x, tid = threadIdx.x;

    unsigned int cnt[8], loc = 0;
    for (int k = 0; k < 8; ++k) {              // rank r = 8*tid+k from the top
        cnt[k] = hist[b * NBIN + (NBIN - 1 - (tid * 8 + k))];
        loc += cnt[k];
    }
    part[tid] = loc;
    if (tid == 0) cutBin = -1;
    __syncthreads();
    for (int off = 1; off < 512; off <<= 1) {   // inclusive scan (Hillis-Steele)
        unsigned int v = (tid >= off) ? part[tid - off] : 0u;
        __syncthreads();
        part[tid] += v;
        __syncthreads();
    }
    unsigned int runb = (tid > 0) ? part[tid - 1] : 0u;   // exclusive prefix
    for (int k = 0; k < 8; ++k) {
        unsigned int runa = runb + cnt[k];
        if (runb < PRE_K && runa >= PRE_K) cutBin = NBIN - 1 - (tid * 8 + k);
        runb = runa;
    }
    __syncthreads();
    if (tid == 0) {
        int cb = cutBin;
        if (part[511] < PRE_K) cb = 0;          // fewer than PRE_K total: take all
        meta[b] = cb;
    }
}

// ---------------------------------------------------------------------------
// Pass 2: recompute scores (bit-identical) and gather candidates.
__global__ __launch_bounds__(256) void k_gather(Ins ins, const int* meta,
                                                float* candS, int* candI,
                                                float* boundS, int* boundI,
                                                unsigned int* ctrA, unsigned int* ctrB) {
    const int b = blockIdx.y;
    const int g = blockIdx.x * blockDim.x + threadIdx.x;
    if (g >= NANCH) return;
    const int cut = meta[b];

    int lvl, local, hw; float stride;
    resolveAnchor(g, lvl, local, hw, stride);
    const int a   = local / hw;
    const int rem = local - a * hw;
    const float* inp = ins.in[lvl];
    const int base = ((b * 3 + a) * (NUM_CLASSES + 5)) * hw + rem;

    const float conf = sigm(inp[base + 4 * hw]);
    for (int c = 0; c < NUM_CLASSES; ++c) {
        const float s = conf * sigm(inp[base + (5 + c) * hw]);
        if (s > SCORE_TH) {
            const int bin = scoreBin(s);
            const int id  = g * NUM_CLASSES + c;
            if (bin > cut) {
                unsigned int p = atomicAdd(&ctrA[b], 1u);
                if (p < PRE_K) { candS[b * PRE_K + p] = s; candI[b * PRE_K + p] = id; }
            } else if (bin == cut) {
                unsigned int p = atomicAdd(&ctrB[b], 1u);
                if (p < BOUND_CAP) { boundS[b * BOUND_CAP + p] = s; boundI[b * BOUND_CAP + p] = id; }
            }
        }
    }
}

// ---------------------------------------------------------------------------
// Exact top-PRE_K: bisection on float bits within the boundary bin, compaction,
// padding. Builds the 80KB candidate tile (x1,y1,x2,y2,score SoA planes) in LDS
// and TDM-stores it to global in one DMA.
__global__ __launch_bounds__(512) void k_refine(const float4* boxesWS,
                                                const float* candS, int* candI,
                                                const float* boundS, const int* boundI,
                                                const unsigned int* ctrA, const unsigned int* ctrB,
                                                float* tileWS) {
    extern __shared__ float tile[];               // TILE_NE floats = 80KB (dynamic LDS)
    __shared__ float sS[PRE_K];
    __shared__ int   sI[PRE_K];
    __shared__ unsigned int red[512];
    __shared__ unsigned int posCtr, tieCtr;
    const int b = blockIdx.x, tid = threadIdx.x;

    const unsigned int nA = min(ctrA[b], (unsigned int)PRE_K);
    const unsigned int nB = min(ctrB[b], (unsigned int)BOUND_CAP);
    const int need = PRE_K - (int)nA;
    const float* bsc = boundS + b * BOUND_CAP;
    const int*   bid = boundI + b * BOUND_CAP;

    for (int m = tid; m < (int)nA; m += 512) {
        sS[m] = candS[b * PRE_K + m];
        sI[m] = candI[b * PRE_K + m];
    }
    __syncthreads();

    int M;
    if ((int)nB <= need) {
        for (int m = tid; m < (int)nB; m += 512) { sS[nA + m] = bsc[m]; sI[nA + m] = bid[m]; }
        M = (int)nA + (int)nB;
    } else {
        // bisection: minimal T with #{key > T} < need  ->  V = key of need-th largest
        unsigned int lo = 0u, hi = 0x7F800000u;
        while (lo < hi) {
            const unsigned int mid = (lo + hi) >> 1;
            unsigned int c = 0;
            for (int m = tid; m < (int)nB; m += 512)
                if (__float_as_uint(bsc[m]) > mid) ++c;
            red[tid] = c; __syncthreads();
            for (int off = 256; off; off >>= 1) { if (tid < off) red[tid] += red[tid + off]; __syncthreads(); }
            const unsigned int tot = red[0]; __syncthreads();
            if (tot < (unsigned int)need) hi = mid; else lo = mid + 1;
        }
        const unsigned int V = lo;
        unsigned int c = 0;
        for (int m = tid; m < (int)nB; m += 512)
            if (__float_as_uint(bsc[m]) > V) ++c;
        red[tid] = c; __syncthreads();
        for (int off = 256; off; off >>= 1) { if (tid < off) red[tid] += red[tid + off]; __syncthreads(); }
        const unsigned int cntGt = red[0]; __syncthreads();

        if (tid == 0) { posCtr = 0u; tieCtr = 0u; }
        __syncthreads();
        const unsigned int tieTake = (unsigned int)need - cntGt;
        for (int m = tid; m < (int)nB; m += 512) {
            const unsigned int k = __float_as_uint(bsc[m]);
            if (k > V) {
                const unsigned int p = atomicAdd(&posCtr, 1u);
                sS[nA + p] = bsc[m]; sI[nA + p] = bid[m];
            } else if (k == V) {
                const unsigned int t = atomicAdd(&tieCtr, 1u);
                if (t < tieTake) { sS[nA + cntGt + t] = bsc[m]; sI[nA + cntGt + t] = bid[m]; }
            }
        }
        M = PRE_K;
    }
    __syncthreads();
    for (int m = M + tid; m < PRE_K; m += 512) { sS[m] = NEGV; sI[m] = 0; }
    __syncthreads();

    for (int m = tid; m < PRE_K; m += 512) {
        const float s = sS[m];
        const int  id = sI[m];
        candI[b * PRE_K + m] = id;
        float4 bb = make_float4(0.f, 0.f, 0.f, 0.f);
        float off = 0.f;
        if (s > NEGV * 0.5f) {
            const int anc = id / NUM_CLASSES;
            const int cls = id - anc * NUM_CLASSES;
            bb  = boxesWS[b * NANCH + anc];
            off = (float)cls * 4096.0f;              // class-aware NMS offset
        }
        tile[m]              = bb.x + off;
        tile[PRE_K + m]      = bb.y + off;
        tile[2 * PRE_K + m]  = bb.z + off;
        tile[3 * PRE_K + m]  = bb.w + off;
        tile[4 * PRE_K + m]  = s;
    }
    __syncthreads();                              // all ds-writes complete

    // ---- TDM: 80KB LDS -> global, one DMA issued by wave 0 ----
    if (tid < 32) {
        u32x4v g0; i32x8v g1;
        i32x4v g2 = { 0, 0, 0, 0 };
        i32x4v g3 = { 0, 0, 0, 0 };
        i32x8v g4 = { 0, 0, 0, 0, 0, 0, 0, 0 };
        make_tdm_desc((unsigned int)(size_t)(void*)&tile[0],
                      (unsigned long long)(size_t)(tileWS + (size_t)b * TILE_NE),
                      TILE_NE, g0, g1);
        __builtin_amdgcn_tensor_store_from_lds(g0, g1, g2, g3, g4, 0);
        __builtin_amdgcn_s_wait_tensorcnt(0);
    }
}

// ---------------------------------------------------------------------------
// Greedy NMS, one block per image. Candidate tile (boxes+scores, 80KB) is
// DMA'd into LDS by the Tensor Data Mover and fenced with s_wait_tensorcnt.
__global__ __launch_bounds__(512) void k_nms(const int* candI, const float* tileWS,
                                             const float4* boxesWS, float* out) {
    extern __shared__ float ob[];                 // TILE_NE floats = 80KB (dynamic LDS)
    __shared__ float rS[16];
    __shared__ int   rI[16];
    __shared__ int   wIx;
    __shared__ float selS[TOP_K];
    __shared__ int   selI[TOP_K];

    const int b = blockIdx.x, tid = threadIdx.x;
    const int lane = tid & 31, wid = tid >> 5;

    // ---- TDM: 80KB global -> LDS, issued by wave 0 (EXEC all-ones) ----
    if (tid < 32) {
        u32x4v g0; i32x8v g1;
        i32x4v g2 = { 0, 0, 0, 0 };
        i32x4v g3 = { 0, 0, 0, 0 };
        i32x8v g4 = { 0, 0, 0, 0, 0, 0, 0, 0 };
        make_tdm_desc((unsigned int)(size_t)(void*)&ob[0],
                      (unsigned long long)(size_t)(tileWS + (size_t)b * TILE_NE),
                      TILE_NE, g0, g1);
        __builtin_amdgcn_tensor_load_to_lds(g0, g1, g2, g3, g4, 0);
        __builtin_amdgcn_s_wait_tensorcnt(0);
    }
    __syncthreads();

    // Per-thread candidates j = k*512 + tid (k < 8): coords/areas/scores in regs.
    float s[8], bx1[8], by1[8], bx2[8], by2[8], ar[8];
    for (int k = 0; k < 8; ++k) {
        const int j = k * 512 + tid;
        bx1[k] = ob[j];
        by1[k] = ob[PRE_K + j];
        bx2[k] = ob[2 * PRE_K + j];
        by2[k] = ob[3 * PRE_K + j];
        s[k]   = ob[4 * PRE_K + j];
        ar[k]  = (bx2[k] - bx1[k]) * (by2[k] - by1[k]);
    }

    for (int t = 0; t < TOP_K; ++t) {
        // argmax over live scores: regs -> wave32 shuffle -> 16-wave LDS reduce
        float best = -3.0e38f; int bidx = tid;
        for (int k = 0; k < 8; ++k)
            if (s[k] > best) { best = s[k]; bidx = k * 512 + tid; }
        for (int off = 16; off; off >>= 1) {
            const float os = __shfl_down(best, off, 32);
            const int   oi = __shfl_down(bidx, off, 32);
            if (os > best) { best = os; bidx = oi; }
        }
        if (lane == 0) { rS[wid] = best; rI[wid] = bidx; }
        __syncthreads();
        if (tid < 16) {
            best = rS[tid]; bidx = rI[tid];
            for (int off = 8; off; off >>= 1) {
                const float os = __shfl_down(best, off, 16);
                const int   oi = __shfl_down(bidx, off, 16);
                if (os > best) { best = os; bidx = oi; }
            }
            if (tid == 0) { wIx = bidx; selS[t] = best; selI[t] = bidx; }
        }
        __syncthreads();

        // suppress vs winner (IoU in offset space; self-suppression included)
        const int wi = wIx;
        const float wx1 = ob[wi], wy1 = ob[PRE_K + wi];
        const float wx2 = ob[2 * PRE_K + wi], wy2 = ob[3 * PRE_K + wi];
        const float wa  = (wx2 - wx1) * (wy2 - wy1);
        for (int k = 0; k < 8; ++k) {
            const float ix1 = fmaxf(wx1, bx1[k]), iy1 = fmaxf(wy1, by1[k]);
            const float ix2 = fminf(wx2, bx2[k]), iy2 = fminf(wy2, by2[k]);
            const float inter = fmaxf(ix2 - ix1, 0.0f) * fmaxf(iy2 - iy1, 0.0f);
            const float iou   = inter / (wa + ar[k] - inter + 1e-9f);
            if (iou > NMS_TH) s[k] = NEGV;
        }
        __syncthreads();
    }

    if (tid < TOP_K) {
        const float sc = selS[tid];
        const int   i  = selI[tid];
        float o0 = 0.f, o1 = 0.f, o2 = 0.f, o3 = 0.f, o4 = 0.f, o5 = 0.f;
        if (sc > SCORE_TH) {
            const int id  = candI[b * PRE_K + i];
            const int anc = id / NUM_CLASSES;
            const int cls = id - anc * NUM_CLASSES;
            const float4 bb = boxesWS[b * NANCH + anc];
            o0 = bb.x; o1 = bb.y; o2 = bb.z; o3 = bb.w; o4 = (float)cls; o5 = sc;
        }
        float* o = out + ((size_t)b * TOP_K + tid) * 6;
        o[0] = o0; o[1] = o1; o[2] = o2; o[3] = o3; o[4] = o4; o[5] = o5;
    }
}

// ---------------------------------------------------------------------------
extern "C" void kernel_launch(void* const* d_in, const int* in_sizes, int n_in,
                              void* d_out, int out_size, void* d_ws, size_t ws_size,
                              hipStream_t stream) {
    (void)in_sizes; (void)n_in; (void)out_size; (void)ws_size;

    Ins ins;
    for (int l = 0; l < 3; ++l) {
        ins.in[l] = (const float*)d_in[l * 5 + 0];
        ins.aw[l] = (const float*)d_in[l * 5 + 1];
        ins.ah[l] = (const float*)d_in[l * 5 + 2];
        ins.gx[l] = (const float*)d_in[l * 5 + 3];
        ins.gy[l] = (const float*)d_in[l * 5 + 4];
    }

    char* w = (char*)d_ws;
    size_t off = 0;
    auto take = [&](size_t bytes) -> char* {
        char* r = w + off;
        off += (bytes + 255) & ~(size_t)255;
        return r;
    };
    float4*       boxesWS = (float4*)      take((size_t)BS * NANCH * sizeof(float4));   // 2.91 MB
    unsigned int* hist    = (unsigned int*)take((size_t)BS * NBIN * 4);                 // 128 KB
    unsigned int* ctrA    = (unsigned int*)take(BS * 4);
    unsigned int* ctrB    = (unsigned int*)take(BS * 4);
    int*          meta    = (int*)         take(BS * 4);
    float*        candS   = (float*)       take((size_t)BS * PRE_K * 4);
    int*          candI   = (int*)         take((size_t)BS * PRE_K * 4);
    float*        boundS  = (float*)       take((size_t)BS * BOUND_CAP * 4);
    int*          boundI  = (int*)         take((size_t)BS * BOUND_CAP * 4);
    float*        tileWS  = (float*)       take((size_t)BS * TILE_NE * 4);              // 640 KB

    k_init<<<dim3((BS * NBIN + 255) / 256), 256, 0, stream>>>(hist, ctrA, ctrB);

    dim3 gAnch((NANCH + 255) / 256, BS);
    k_decode_hist<<<gAnch, 256, 0, stream>>>(ins, boxesWS, hist);
    k_scan<<<BS, 512, 0, stream>>>(hist, meta);
    k_gather<<<gAnch, 256, 0, stream>>>(ins, meta, candS, candI, boundS, boundI, ctrA, ctrB);
    k_refine<<<BS, 512, (size_t)TILE_NE * sizeof(float), stream>>>(boxesWS, candS, candI,
                                                                   boundS, boundI, ctrA, ctrB,
                                                                   tileWS);
    k_nms<<<BS, 512, (size_t)TILE_NE * sizeof(float), stream>>>(candI, tileWS, boxesWS,
                                                                (float*)d_out);
}